// Code2seqTokEmbedWithVal_54855322304735
// MI455X (gfx1250) — compile-verified
//
#include <hip/hip_runtime.h>
#include <hip/hip_bf16.h>

// ---------------------------------------------------------------------------
// Problem shape (from reference): ROWS = 131072, E = 256, NNZ = 524288,
// NUM_NODE_TYPES = 128, NUM_VAL_TOKENS = 50000.
// Output: [node_embed (ROWS*E f32)] ++ [node_val_embed (ROWS*E f32)]
// Memory-bound problem (~0.27 GFLOP vs ~0.6 GB moved -> ~30us at 23.3 TB/s);
// CDNA5-specific path: async global->LDS staging + ds_load_b128 gathers.
// ---------------------------------------------------------------------------

typedef float f4 __attribute__((ext_vector_type(4)));
typedef int v4i __attribute__((vector_size(16)));   // matches builtin param type

#define AS1 __attribute__((address_space(1)))
#define AS3 __attribute__((address_space(3)))

// Async global -> LDS copy of 16 bytes (CDNA5 GLOBAL_LOAD_ASYNC_TO_LDS_B128,
// tracked with ASYNCcnt). Builtin signature: (v4i AS1*, v4i AS3*, Ii, Ii).
__device__ __forceinline__ void async_g2l_b128(const void* g, void* l) {
#if defined(__HIP_DEVICE_COMPILE__)
#if __has_builtin(__builtin_amdgcn_global_load_async_to_lds_b128)
  __builtin_amdgcn_global_load_async_to_lds_b128(
      (AS1 v4i*)g, (AS3 v4i*)l, /*offset=*/0, /*cpol=*/0);
#else
  unsigned loff = (unsigned)(unsigned long long)(AS3 void*)l;
  asm volatile("global_load_async_to_lds_b128 %0, %1, off"
               :
               : "v"(loff), "v"(g)
               : "memory");
#endif
#endif
}

__device__ __forceinline__ void wait_async0() {
#if defined(__HIP_DEVICE_COMPILE__)
#if __has_builtin(__builtin_amdgcn_s_wait_asynccnt)
  __builtin_amdgcn_s_wait_asynccnt(0);
#else
  asm volatile("s_wait_asynccnt 0" ::: "memory");
#endif
#endif
}

// ---------------------------------------------------------------------------
// Kernel 1: node_embed = node_embed_w[node_idx]
// Stage the whole 128x256 f32 table (128 KB) into LDS via async loads, then
// stream rows_per_block rows per block: ds_load_b128 -> global_store_b128.
// Thread t: sub-row r = t>>6 (4 rows in flight), f4 slot q = t&63.
// ---------------------------------------------------------------------------
__global__ void __launch_bounds__(256) node_embed_gather_kernel(
    const int* __restrict__ node_idx,
    const float* __restrict__ w,   // ntypes x 256
    float* __restrict__ out,       // nrows x 256
    int nrows, int rows_per_block, int n_f4 /* = ntypes*64 */) {
  extern __shared__ float lut[];
  const int t = threadIdx.x;

  const f4* gsrc = (const f4*)w;
  f4* ldst = (f4*)lut;
  for (int i = t; i < n_f4; i += 256) {
    async_g2l_b128((const void*)(gsrc + i), (void*)(ldst + i));
  }
  wait_async0();
  __syncthreads();

  const int r = t >> 6;   // 0..3
  const int q = t & 63;   // f4 index within a 256-float row
  const int row0 = blockIdx.x * rows_per_block;
  for (int rr = r; rr < rows_per_block; rr += 4) {
    const int row = row0 + rr;
    if (row >= nrows) break;
    const int idx = node_idx[row];
    f4 v = ((const f4*)lut)[idx * 64 + q];          // ds_load_b128
    ((f4*)out)[(long long)row * 64 + q] = v;        // global_store_b128
  }
}

// ---------------------------------------------------------------------------
// Kernel 2: node_val_embed[row] = sum_j (rows[j]==row) vals[j]*vt[cols[j]]
// rows[] is sorted -> each output row owns a contiguous nnz range; one wave32
// per row finds it with two binary searches (wave-uniform -> scalar loads),
// then lane L accumulates columns [4L..4L+3] and [128+4L..128+4L+3].
// Deterministic: no atomics; empty rows write zeros.
// ---------------------------------------------------------------------------
__global__ void __launch_bounds__(256) spmm_segsum_kernel(
    const int* __restrict__ rows_arr,
    const int* __restrict__ cols,
    const float* __restrict__ vals,
    const float* __restrict__ vt,   // ntok x 256
    float* __restrict__ out,        // nrows x 256
    int nrows, int nnz) {
  const int lane = threadIdx.x & 31;
  const int wid = blockIdx.x * (blockDim.x >> 5) + (threadIdx.x >> 5);
  const int row = __builtin_amdgcn_readfirstlane(wid);
  if (row >= nrows) return;

  // lower_bound(row) over sorted rows_arr
  int lo = 0, hi = nnz;
  while (lo < hi) {
    const int mid = (lo + hi) >> 1;
    if (rows_arr[mid] < row) lo = mid + 1; else hi = mid;
  }
  const int start = lo;
  // lower_bound(row+1), starting from start
  hi = nnz;
  while (lo < hi) {
    const int mid = (lo + hi) >> 1;
    if (rows_arr[mid] < row + 1) lo = mid + 1; else hi = mid;
  }
  const int end = lo;

  f4 acc0 = {0.f, 0.f, 0.f, 0.f};
  f4 acc1 = {0.f, 0.f, 0.f, 0.f};
  for (int j = start; j < end; ++j) {
    const int col = cols[j];       // wave-uniform
    const float v = vals[j];       // wave-uniform
    const f4* p = (const f4*)vt + (long long)col * 64;
    const f4 a = p[lane];          // global_load_b128, 512B coalesced
    const f4 b = p[32 + lane];     // global_load_b128, 512B coalesced
    acc0 += a * v;
    acc1 += b * v;
  }

  f4* o = (f4*)out + (long long)row * 64;
  o[lane] = acc0;
  o[32 + lane] = acc1;
}

// ---------------------------------------------------------------------------
// Launch. Inputs (setup_inputs order):
//   0 node_idx   (S*N*B = 131072 int)
//   1 spmm_rows  (NNZ int, sorted)
//   2 spmm_cols  (NNZ int)
//   3 spmm_vals  (NNZ f32)
//   4 node_embed_w  (128*256 f32)
//   5 val_tok_embed (50000*256 f32)
// ---------------------------------------------------------------------------
extern "C" void kernel_launch(void* const* d_in, const int* in_sizes, int n_in,
                              void* d_out, int out_size, void* d_ws, size_t ws_size,
                              hipStream_t stream) {
  const int*   node_idx = (const int*)d_in[0];
  const int*   rows_arr = (const int*)d_in[1];
  const int*   cols     = (const int*)d_in[2];
  const float* vals     = (const float*)d_in[3];
  const float* w        = (const float*)d_in[4];
  const float* vt       = (const float*)d_in[5];
  float*       out      = (float*)d_out;

  const int E      = 256;
  const int nrows  = in_sizes[0];            // 131072
  const int nnz    = in_sizes[1];            // 524288
  const int ntypes = in_sizes[4] / E;        // 128
  const int n_f4   = ntypes * (E / 4);       // 8192 f4 = 128 KB

  // Part 1: dense embedding gather (LDS-staged table).
  const int rows_per_block = 128;
  const int blocks1 = (nrows + rows_per_block - 1) / rows_per_block;
  const size_t lds_bytes = (size_t)ntypes * E * sizeof(float);  // 131072 B
  node_embed_gather_kernel<<<blocks1, 256, lds_bytes, stream>>>(
      node_idx, w, out, nrows, rows_per_block, n_f4);

  // Part 2: sorted-COO SpMM segment-sum; one wave per output row.
  float* out2 = out + (long long)nrows * E;
  const int waves_per_block = 8;             // 256 threads, wave32
  const int blocks2 = (nrows + waves_per_block - 1) / waves_per_block;
  spmm_segsum_kernel<<<blocks2, 256, 0, stream>>>(
      rows_arr, cols, vals, vt, out2, nrows, nnz);
}